// SLANeXtSLAHead_70540542869508
// MI455X (gfx1250) — compile-verified
//
#include <hip/hip_runtime.h>
#include <hip/hip_bf16.h>
#include <stdint.h>

// ---------------- problem constants ----------------
#define BB      32
#define NN      640
#define CC      512
#define HH      512
#define NE      30
#define NSTEPS  201
#define NSLICE  4
#define NROWS   (NN / NSLICE)     // 160 rows of feat_proj per slice-block
#define TPB     512               // decode kernel threads (16 waves)
#define NBLK    (BB * NSLICE)     // 128 persistent blocks
#define KX      (CC + NE)         // 542 = GRU input width
#define KXP     544               // padded to multiple of 4
#define XS_STR  548               // LDS row stride for x tile (bank-conflict free)
#define BS_STR  516               // LDS row stride for GEMM B panel

// ---------------- workspace layout (floats) ----------------
#define OFF_FP    0u                           // feat_proj [B*N, H]   10485760
#define OFF_CTXP  10485760u                    // ctx partials [B][4][C]  65536
#define OFF_MZ    (OFF_CTXP + 65536u)          // (m,z) per [B][4]          256
#define OFF_H     (OFF_MZ + 256u)              // h double buffer 2*B*H   32768
#define OFF_WEFF  (OFF_H + 32768u)             // folded W_s2@W_s1        15360
#define OFF_BEFF  (OFF_WEFF + 15360u)          // folded bias                32
#define OFF_PRE   (OFF_BEFF + 32u)             // pre chars (int)            32
#define OFF_BAR   (OFF_PRE + 32u)              // barrier counters (u32)      8

typedef float v2f __attribute__((ext_vector_type(2)));
typedef float v8f __attribute__((ext_vector_type(8)));

__device__ __forceinline__ v8f wmma_f32_k4(v2f a, v2f b, v8f c) {
  // full-precision fp32 WMMA: D = A(16x4) * B(4x16) + C
  return __builtin_amdgcn_wmma_f32_16x16x4_f32(false, a, false, b, (short)0, c,
                                               false, false);
}

__device__ __forceinline__ float sigm(float x) {
  return 1.0f / (1.0f + __expf(-x));
}

// CDNA5 hardware V_TANH_F32 (TRANS32) when the toolchain exposes it; the
// score loop evaluates ~10.5M tanh per step, so this is the hot VALU path.
__device__ __forceinline__ float fast_tanh(float x) {
#if __has_builtin(__builtin_amdgcn_tanhf)
  return __builtin_amdgcn_tanhf(x);
#else
  return tanhf(x);
#endif
}

// ---------------- device-wide sense barrier (all NBLK blocks resident) ------
__device__ __forceinline__ void grid_sync(unsigned* bar, unsigned expect) {
  __syncthreads();
  if (threadIdx.x == 0) {
    __threadfence();
    unsigned prev = __hip_atomic_fetch_add(&bar[0], 1u, __ATOMIC_ACQ_REL,
                                           __HIP_MEMORY_SCOPE_AGENT);
    if (prev == (unsigned)(NBLK - 1)) {
      __hip_atomic_store(&bar[0], 0u, __ATOMIC_RELAXED, __HIP_MEMORY_SCOPE_AGENT);
      __hip_atomic_store(&bar[1], expect, __ATOMIC_RELEASE, __HIP_MEMORY_SCOPE_AGENT);
    } else {
      while (__hip_atomic_load(&bar[1], __ATOMIC_ACQUIRE,
                               __HIP_MEMORY_SCOPE_AGENT) < expect)
        __builtin_amdgcn_s_sleep(2);
    }
    __threadfence();
  }
  __syncthreads();
}

// ---------------- init: zero barrier/pre/h0 --------------------------------
__global__ void init_kernel(float* hbuf, int* pre, unsigned* bar) {
  const int tid = threadIdx.x;
  for (int i = tid; i < BB * HH; i += 256) hbuf[i] = 0.0f;  // h parity-0 buffer
  if (tid < BB) pre[tid] = 0;
  if (tid < 8) bar[tid] = 0u;
}

// ---------------- fold structure MLP: Weff = W_s2 @ W_s1 -------------------
__global__ void weff_kernel(const float* __restrict__ Ws1,
                            const float* __restrict__ bs1,
                            const float* __restrict__ Ws2,
                            const float* __restrict__ bs2,
                            float* __restrict__ Weff, float* __restrict__ beff) {
  const int idx = blockIdx.x * 256 + threadIdx.x;
  if (idx < NE * HH) {
    const int e = idx >> 9, h = idx & 511;
    float a = 0.0f;
    for (int k = 0; k < HH; ++k) a += Ws2[(e << 9) + k] * Ws1[(k << 9) + h];
    Weff[idx] = a;
  }
  if (blockIdx.x == 0 && threadIdx.x < NE) {
    const int e = threadIdx.x;
    float a = bs2[e];
    for (int k = 0; k < HH; ++k) a += Ws2[(e << 9) + k] * bs1[k];
    beff[e] = a;
  }
}

// ---------------- feat_proj = features @ W_i2h^T  (fp32 WMMA GEMM) ---------
// grid = (160, 32), 256 threads (8 waves). Each wave owns one 16x16 tile.
// B panel (16 cols x K=512 of W_i2h) staged to LDS with async-to-LDS copies.
__global__ __launch_bounds__(256) void featproj_kernel(
    const float* __restrict__ features, const float* __restrict__ W_i2h,
    float* __restrict__ fp) {
  __shared__ float Bs[16 * BS_STR];

  const int tid  = threadIdx.x;
  const int col0 = blockIdx.y * 16;

  // async stage: 16 rows x 512 floats = 2048 16B chunks, 256 threads x 8 iters
#pragma unroll
  for (int it = 0; it < 8; ++it) {
    const int cid = it * 256 + tid;
    const int row = cid >> 7;          // 0..15
    const int cc  = cid & 127;         // 16B chunk within row
    const float* g = W_i2h + (size_t)(col0 + row) * CC + cc * 4;
    const uint32_t loff = (uint32_t)(uintptr_t)(&Bs[row * BS_STR + cc * 4]);
    asm volatile("global_load_async_to_lds_b128 %0, %1, off"
                 :: "v"(loff), "v"(g) : "memory");
  }
  asm volatile("s_wait_asynccnt 0" ::: "memory");
  __syncthreads();

  const int w  = tid >> 5, ln = tid & 31;
  const int rt = blockIdx.x * 8 + w;               // row tile 0..1279
  const int mm = ln & 15;                          // frag m (A) / n (B)
  const int kb = (ln >> 4) << 1;                   // k-pair select

  const float* Arow = features + (size_t)(rt * 16 + mm) * CC;
  v8f acc0 = {}, acc1 = {};
#pragma unroll 4
  for (int k0 = 0; k0 < CC; k0 += 8) {
    {
      const int k = k0 + kb;
      v2f a, b;
      a.x = Arow[k];               a.y = Arow[k + 1];
      b.x = Bs[mm * BS_STR + k];   b.y = Bs[mm * BS_STR + k + 1];
      acc0 = wmma_f32_k4(a, b, acc0);
    }
    {
      const int k = k0 + 4 + kb;
      v2f a, b;
      a.x = Arow[k];               a.y = Arow[k + 1];
      b.x = Bs[mm * BS_STR + k];   b.y = Bs[mm * BS_STR + k + 1];
      acc1 = wmma_f32_k4(a, b, acc1);
    }
  }
  const int mrow = (ln >= 16) ? 8 : 0;
  float* drow = fp + (size_t)(rt * 16) * HH + col0;
#pragma unroll
  for (int i = 0; i < 8; ++i)
    drow[(size_t)(mrow + i) * HH + mm] = acc0[i] + acc1[i];
}

// ---------------- structure-head epilogue (per batch element) --------------
__device__ __forceinline__ void structure_epilogue(
    int b, const float* __restrict__ hrow, int tOut,
    const float* __restrict__ Weff, const float* __restrict__ beff,
    float* __restrict__ out, int* __restrict__ pre, float* lg, int tid) {
  const int w = tid >> 5, ln = tid & 31;
  if (w < 15) {  // waves 0..14 cover e = 0..29 (2 each)
#pragma unroll
    for (int ee = 0; ee < 2; ++ee) {
      const int e = w * 2 + ee;
      const float* wr = Weff + (e << 9);
      float a = 0.0f;
#pragma unroll
      for (int i = 0; i < 16; ++i) {
        const int h = ln + (i << 5);
        a += hrow[h] * wr[h];
      }
#pragma unroll
      for (int o = 16; o; o >>= 1) a += __shfl_xor(a, o, 32);
      if (ln == 0) lg[e] = a + beff[e];
    }
  }
  __syncthreads();
  if (w == 0) {
    const float v = (ln < NE) ? lg[ln] : -3.4e38f;
    float mv = v;
    int mi = ln;
#pragma unroll
    for (int o = 16; o; o >>= 1) {  // argmax, first-max tie-break
      const float ov = __shfl_xor(mv, o, 32);
      const int oi = __shfl_xor(mi, o, 32);
      if (ov > mv || (ov == mv && oi < mi)) { mv = ov; mi = oi; }
    }
    float ex = (ln < NE) ? __expf(v - mv) : 0.0f;
    float ss = ex;
#pragma unroll
    for (int o = 16; o; o >>= 1) ss += __shfl_xor(ss, o, 32);
    if (ln < NE) out[((size_t)b * NSTEPS + tOut) * NE + ln] = ex / ss;
    if (ln == 0) pre[b] = mi;
  }
  __syncthreads();
}

// ---------------- persistent decode kernel ---------------------------------
// 128 blocks resident: block = (b = id&31, slice = id>>5). 2 barriers/step.
__global__ __launch_bounds__(TPB) void decode_kernel(
    const float* __restrict__ fp, const float* __restrict__ features,
    const float* __restrict__ W_h2h, const float* __restrict__ b_h2h,
    const float* __restrict__ w_score, const float* __restrict__ W_ih,
    const float* __restrict__ W_hh, const float* __restrict__ b_ih,
    const float* __restrict__ b_hh, const float* __restrict__ Weff,
    const float* __restrict__ beff, float* __restrict__ ctxp,
    float* __restrict__ mz, float* __restrict__ hbuf, int* __restrict__ pre,
    unsigned* __restrict__ bar, float* __restrict__ out) {
  __shared__ float qs[HH];
  __shared__ float wsc[HH];
  __shared__ float sc[NROWS];
  __shared__ float red0;
  __shared__ float xs[32 * XS_STR];
  __shared__ float esm[BB * NSLICE];
  __shared__ int pint[BB];
  __shared__ float gsm[6 * 32 * 16];
  __shared__ float lg[32];

  const int tid = threadIdx.x;
  const int bid = blockIdx.x;
  const int b = bid & 31;
  const int s = bid >> 5;
  const int nbase = s * NROWS;
  const int wv = tid >> 5, ln = tid & 31;

  // loop-invariant staging
  for (int i = tid; i < HH; i += TPB) wsc[i] = w_score[i];
  __syncthreads();

  unsigned expect = 0;
  for (int t = 0; t < NSTEPS; ++t) {
    const float* hcur = hbuf + (size_t)(t & 1) * BB * HH;      // h_{t-1}
    float* hnext = hbuf + (size_t)((t + 1) & 1) * BB * HH;     // h_t

    // ---- logits/argmax/softmax for previous step (slice 0 only) ----
    if (s == 0 && t > 0)
      structure_epilogue(b, hcur + (b << 9), t - 1, Weff, beff, out, pre, lg, tid);

    // ---- q = h @ W_h2h^T + b_h2h  (one output per thread) ----
    {
      const int h = tid;
      float acc = b_h2h[h];
      const float* wr = W_h2h + (h << 9);
      const float* hv = hcur + (b << 9);
#pragma unroll 4
      for (int k = 0; k < HH; k += 4) {
        const float4 wv4 = *(const float4*)(wr + k);
        const float4 hv4 = *(const float4*)(hv + k);
        acc += wv4.x * hv4.x + wv4.y * hv4.y + wv4.z * hv4.z + wv4.w * hv4.w;
      }
      qs[h] = acc;
    }
    __syncthreads();

    // ---- scores[n] = w_score . tanh(feat_proj[n,:] + q) ----
    for (int n = wv; n < NROWS; n += 16) {
      const float* fpr = fp + ((size_t)(b * NN + nbase + n)) * HH;
      float a = 0.0f;
#pragma unroll
      for (int i = 0; i < 16; ++i) {
        const int h = ln + (i << 5);
        a += fast_tanh(fpr[h] + qs[h]) * wsc[h];
      }
#pragma unroll
      for (int o = 16; o; o >>= 1) a += __shfl_xor(a, o, 32);
      if (ln == 0) sc[n] = a;
    }
    __syncthreads();

    // ---- online-softmax partial (m, z) for this slice ----
    if (wv == 0) {
      float m = -3.4e38f;
      for (int n = ln; n < NROWS; n += 32) m = fmaxf(m, sc[n]);
#pragma unroll
      for (int o = 16; o; o >>= 1) m = fmaxf(m, __shfl_xor(m, o, 32));
      float zz = 0.0f;
      for (int n = ln; n < NROWS; n += 32) zz += __expf(sc[n] - m);
#pragma unroll
      for (int o = 16; o; o >>= 1) zz += __shfl_xor(zz, o, 32);
      if (ln == 0) {
        red0 = m;
        mz[((b * NSLICE + s) << 1) + 0] = m;
        mz[((b * NSLICE + s) << 1) + 1] = zz;
      }
    }
    __syncthreads();
    for (int n = tid; n < NROWS; n += TPB) sc[n] = __expf(sc[n] - red0);
    __syncthreads();

    // ---- unnormalized partial context: ctxp[c] = sum_n p[n]*features ----
    {
      const int c = tid;  // TPB == CC
      const float* fb = features + ((size_t)(b * NN + nbase)) * CC + c;
      float a = 0.0f;
#pragma unroll 4
      for (int n = 0; n < NROWS; ++n) a += sc[n] * fb[(size_t)n * CC];
      ctxp[((b * NSLICE + s) << 9) + c] = a;
    }

    grid_sync(bar, ++expect);  // ---- barrier 1: ctx partials + pre ready ----

    // ---- GRU gates via fp32 WMMA; block j owns h columns [16j,16j+16) ----
    if (bid < 32) {
      const int j = bid;
      if (tid < BB) {  // combine scales per batch element
        const int bb = tid;
        const float m0 = mz[(bb * 4 + 0) * 2], z0 = mz[(bb * 4 + 0) * 2 + 1];
        const float m1 = mz[(bb * 4 + 1) * 2], z1 = mz[(bb * 4 + 1) * 2 + 1];
        const float m2 = mz[(bb * 4 + 2) * 2], z2 = mz[(bb * 4 + 2) * 2 + 1];
        const float m3 = mz[(bb * 4 + 3) * 2], z3 = mz[(bb * 4 + 3) * 2 + 1];
        const float M = fmaxf(fmaxf(m0, m1), fmaxf(m2, m3));
        const float e0 = __expf(m0 - M), e1 = __expf(m1 - M);
        const float e2 = __expf(m2 - M), e3 = __expf(m3 - M);
        const float Z = z0 * e0 + z1 * e1 + z2 * e2 + z3 * e3;
        esm[bb * 4 + 0] = e0 / Z;
        esm[bb * 4 + 1] = e1 / Z;
        esm[bb * 4 + 2] = e2 / Z;
        esm[bb * 4 + 3] = e3 / Z;
        pint[bb] = pre[bb];
      }
      __syncthreads();
      // build x = [ctx | one_hot(pre) | 0pad] tile (32 x 544) in LDS
      for (int idx = tid; idx < 32 * KXP; idx += TPB) {
        const int m = idx / KXP, k = idx - m * KXP;
        float v;
        if (k < CC) {
          v = ctxp[((m * 4 + 0) << 9) + k] * esm[m * 4 + 0] +
              ctxp[((m * 4 + 1) << 9) + k] * esm[m * 4 + 1] +
              ctxp[((m * 4 + 2) << 9) + k] * esm[m * 4 + 2] +
              ctxp[((m * 4 + 3) << 9) + k] * esm[m * 4 + 3];
        } else if (k < KX) {
          v = (k - CC == pint[m]) ? 1.0f : 0.0f;
        } else {
          v = 0.0f;
        }
        xs[m * XS_STR + k] = v;
      }
      __syncthreads();

      // 12 WMMA jobs: {i-part,h-part} x {r,z,g} x {Mtile 0,1}
      if (wv < 12) {
        const int p = wv / 6, rme = wv % 6, g = rme >> 1, mt = rme & 1;
        const int rb = g * HH + j * 16;
        const int mm = ln & 15;
        const int kb = (ln >> 4) << 1;
        v8f acc0 = {}, acc1 = {};
        if (p == 0) {  // gi slice: A = x (LDS), B = W_ih rows, K = 544
          for (int k0 = 0; k0 < KXP; k0 += 8) {
#pragma unroll
            for (int u = 0; u < 2; ++u) {
              const int k = k0 + u * 4 + kb;
              v2f a, bf;
              const float* xp = &xs[(mt * 16 + mm) * XS_STR + k];
              a.x = xp[0];
              a.y = xp[1];
              if (k < KX) {  // guard ragged K (542)
                const float* wp = W_ih + (size_t)(rb + mm) * KX + k;
                bf.x = wp[0];
                bf.y = wp[1];
              } else {
                bf.x = 0.0f;
                bf.y = 0.0f;
              }
              if (u == 0) acc0 = wmma_f32_k4(a, bf, acc0);
              else        acc1 = wmma_f32_k4(a, bf, acc1);
            }
          }
        } else {  // gh slice: A = h (global/L2), B = W_hh rows, K = 512
          for (int k0 = 0; k0 < HH; k0 += 8) {
#pragma unroll
            for (int u = 0; u < 2; ++u) {
              const int k = k0 + u * 4 + kb;
              v2f a, bf;
              const float* hp = hcur + (size_t)(mt * 16 + mm) * HH + k;
              a.x = hp[0];
              a.y = hp[1];
              const float* wp = W_hh + (size_t)(rb + mm) * HH + k;
              bf.x = wp[0];
              bf.y = wp[1];
              if (u == 0) acc0 = wmma_f32_k4(a, bf, acc0);
              else        acc1 = wmma_f32_k4(a, bf, acc1);
            }
          }
        }
        const int mrow = mt * 16 + ((ln >= 16) ? 8 : 0);
        const int n = ln & 15;
#pragma unroll
        for (int i = 0; i < 8; ++i)
          gsm[((p * 3 + g) * 32 + (mrow + i)) * 16 + n] = acc0[i] + acc1[i];
      }
      __syncthreads();

      // elementwise GRU combine for this block's 16 h-columns
      {
        const int m = tid >> 4, n = tid & 15, col = j * 16 + n;
        const float gi_r = gsm[(0 * 32 + m) * 16 + n] + b_ih[col];
        const float gh_r = gsm[(3 * 32 + m) * 16 + n] + b_hh[col];
        const float gi_z = gsm[(1 * 32 + m) * 16 + n] + b_ih[HH + col];
        const float gh_z = gsm[(4 * 32 + m) * 16 + n] + b_hh[HH + col];
        const float gi_g = gsm[(2 * 32 + m) * 16 + n] + b_ih[2 * HH + col];
        const float gh_g = gsm[(5 * 32 + m) * 16 + n] + b_hh[2 * HH + col];
        const float r = sigm(gi_r + gh_r);
        const float z = sigm(gi_z + gh_z);
        const float nn = fast_tanh(gi_g + r * gh_g);
        hnext[m * HH + col] = (1.0f - z) * nn + z * hcur[m * HH + col];
      }
    }

    grid_sync(bar, ++expect);  // ---- barrier 2: h_t ready ----
  }

  // final step's logits -> out[:, 200, :]
  if (s == 0) {
    const float* hfin = hbuf + (size_t)(NSTEPS & 1) * BB * HH;
    structure_epilogue(b, hfin + (b << 9), NSTEPS - 1, Weff, beff, out, pre, lg,
                       tid);
  }
}

// ---------------- host launch ----------------------------------------------
extern "C" void kernel_launch(void* const* d_in, const int* in_sizes, int n_in,
                              void* d_out, int out_size, void* d_ws,
                              size_t ws_size, hipStream_t stream) {
  const float* features = (const float*)d_in[0];
  const float* W_i2h    = (const float*)d_in[1];
  const float* W_h2h    = (const float*)d_in[2];
  const float* b_h2h    = (const float*)d_in[3];
  const float* w_score  = (const float*)d_in[4];
  const float* W_ih     = (const float*)d_in[5];
  const float* W_hh     = (const float*)d_in[6];
  const float* b_ih     = (const float*)d_in[7];
  const float* b_hh     = (const float*)d_in[8];
  const float* W_s1     = (const float*)d_in[9];
  const float* b_s1     = (const float*)d_in[10];
  const float* W_s2     = (const float*)d_in[11];
  const float* b_s2     = (const float*)d_in[12];
  // d_in[13..16] = location MLP: dead code in the reference output, skipped.

  float* ws = (float*)d_ws;
  float* fp    = ws + OFF_FP;
  float* ctxp  = ws + OFF_CTXP;
  float* mz    = ws + OFF_MZ;
  float* hbuf  = ws + OFF_H;
  float* Weff  = ws + OFF_WEFF;
  float* beff  = ws + OFF_BEFF;
  int*   pre   = (int*)(ws + OFF_PRE);
  unsigned* bar = (unsigned*)(ws + OFF_BAR);
  float* out = (float*)d_out;

  init_kernel<<<1, 256, 0, stream>>>(hbuf, pre, bar);
  weff_kernel<<<(NE * HH + 255) / 256, 256, 0, stream>>>(W_s1, b_s1, W_s2, b_s2,
                                                         Weff, beff);
  featproj_kernel<<<dim3(160, 32), 256, 0, stream>>>(features, W_i2h, fp);
  decode_kernel<<<NBLK, TPB, 0, stream>>>(fp, features, W_h2h, b_h2h, w_score,
                                          W_ih, W_hh, b_ih, b_hh, Weff, beff,
                                          ctxp, mz, hbuf, pre, bar, out);
}